// GroupRenderer_61924838474157
// MI455X (gfx1250) — compile-verified
//
#include <hip/hip_runtime.h>
#include <hip/hip_bf16.h>

typedef __attribute__((ext_vector_type(2))) float v2f;
typedef __attribute__((ext_vector_type(8))) float v8f;

#define LS 128          // L*S flattened feature width
#define NORM_EPS 1e-12f

// ---------------------------------------------------------------- zero output
__global__ __launch_bounds__(256) void gr_zero_kernel(float4* __restrict__ out,
                                                      long long n4) {
  long long i = (long long)blockIdx.x * blockDim.x + threadIdx.x;
  if (i < n4) out[i] = make_float4(0.f, 0.f, 0.f, 0.f);
}

// ------------------------------------------------- WMMA segmented scatter-add
// One wave = one chunk of 16 consecutive samples (sorted ray indices).
// D[m][n] = sum_k A[m][k] * B[k][n] over the 16 samples (4 K-steps of K=4),
// A = weighted one-hot selection (distinct-ray slot x sample), B = group data.
__global__ __launch_bounds__(256) void gr_scatter_wmma_kernel(
    const float* __restrict__ group,
    const float* __restrict__ weights,
    const long long* __restrict__ rays,
    float* __restrict__ acc,
    long long ns, long long nchunks)
{
  const int lane  = threadIdx.x & 31;
  const int tid16 = lane & 15;
  const int half  = lane >> 4;

  long long wave = ((long long)blockIdx.x * blockDim.x + threadIdx.x) >> 5;
  if (wave >= nchunks) return;                  // wave-uniform exit
  const long long base = wave * 16;

  // ---- per-sample metadata: lane (k mod 16) owns sample base+k (both halves)
  long long sm  = base + tid16;
  long long smc = (sm < ns - 1) ? sm : (ns - 1);          // clamp for tail
  int   idx = (int)__builtin_nontemporal_load(&rays[smc]); // rays < 65536
  float w   = 0.f;
  if (sm < ns) w = __builtin_nontemporal_load(&weights[sm]);

  // distinct-segment id via ballot of ray-change flags + prefix popcount.
  // lanes 16..31 hold sample (lane-16), so shfl_up(1) still gives sample-1.
  int prev = __shfl_up(idx, 1, 32);
  bool flag = (tid16 > 0) && (idx != prev);
  unsigned mask = (unsigned)__ballot(flag) & 0xFFFFu;
  int local_id  = __popc(mask & ((1u << tid16) - 1u));     // 0..15
  int ndistinct = __popc(mask) + 1;

  // scatter ray id of each distinct segment into lane 'local_id'
  // (duplicate writers carry identical values -> benign race)
  int ray_of = __builtin_amdgcn_ds_permute(local_id << 2, idx);

  v8f c[8];
  const v8f vzero = {0.f, 0.f, 0.f, 0.f, 0.f, 0.f, 0.f, 0.f};
  #pragma unroll
  for (int t = 0; t < 8; ++t) c[t] = vzero;

  #pragma unroll
  for (int k = 0; k < 4; ++k) {
    // A operand: lane holds A[M=tid16][K=v+2*half], v = vector element
    const int s0 = 4 * k + 2 * half;      // sample offset for v=0 (0..15)
    int   lid0 = __shfl(local_id, s0,     32);
    int   lid1 = __shfl(local_id, s0 + 1, 32);
    float w0   = __shfl(w,        s0,     32);
    float w1   = __shfl(w,        s0 + 1, 32);
    v2f a;
    a.x = (lid0 == tid16) ? w0 : 0.f;
    a.y = (lid1 == tid16) ? w1 : 0.f;

    // B operand: lane holds B[K=v+2*half][N=tid16] = group[sample][16t+tid16]
    long long g0 = base + s0;
    long long g1 = g0 + 1;
    if (g0 > ns - 1) g0 = ns - 1;         // tail clamp (w=0 zeroes contrib)
    if (g1 > ns - 1) g1 = ns - 1;
    const float* p0 = group + g0 * LS + tid16;
    const float* p1 = group + g1 * LS + tid16;

    #pragma unroll
    for (int t = 0; t < 8; ++t) {
      v2f b;
      b.x = __builtin_nontemporal_load(p0 + 16 * t);
      b.y = __builtin_nontemporal_load(p1 + 16 * t);
      // 8 args: (neg_a, A, neg_b, B, c_mod, C, reuse_a, reuse_b)
      c[t] = __builtin_amdgcn_wmma_f32_16x16x4_f32(
          false, a, false, b, (short)0, c[t], false, false);
    }
  }

  // ---- flush: C/D layout puts row M = j + 8*half in VGPR j, col N = tid16.
  // Only rows M < ndistinct carry data; atomics fire at segment granularity.
  #pragma unroll
  for (int j = 0; j < 8; ++j) {
    const int M = j + 8 * half;
    int ray = __shfl(ray_of, M, 32);      // hoisted: all lanes active here
    if (M < ndistinct) {
      float* dst = acc + (long long)ray * LS + tid16;
      #pragma unroll
      for (int t = 0; t < 8; ++t)
        unsafeAtomicAdd(dst + 16 * t, c[t][j]);   // global_atomic_add_f32
    }
  }
}

// --------------------------------------------------- in-place L2 normalize
// one thread per (ray, L) row of 8 contiguous floats
__global__ __launch_bounds__(256) void gr_normalize_kernel(float* __restrict__ out,
                                                           long long nrows) {
  long long i = (long long)blockIdx.x * blockDim.x + threadIdx.x;
  if (i >= nrows) return;
  float4* p = (float4*)(out + i * 8);
  float4 a = p[0];
  float4 b = p[1];
  float ss = a.x * a.x + a.y * a.y + a.z * a.z + a.w * a.w +
             b.x * b.x + b.y * b.y + b.z * b.z + b.w * b.w;
  float s = 1.0f / fmaxf(sqrtf(ss), NORM_EPS);
  a.x *= s; a.y *= s; a.z *= s; a.w *= s;
  b.x *= s; b.y *= s; b.z *= s; b.w *= s;
  p[0] = a;
  p[1] = b;
}

extern "C" void kernel_launch(void* const* d_in, const int* in_sizes, int n_in,
                              void* d_out, int out_size, void* d_ws, size_t ws_size,
                              hipStream_t stream) {
  const float*     group   = (const float*)d_in[0];      // [ns, 16, 8] f32
  const float*     weights = (const float*)d_in[1];      // [ns, 1] f32
  const long long* rays    = (const long long*)d_in[2];  // [ns] int64 (sorted)
  float* out = (float*)d_out;                            // [num_rays, 16, 8] f32

  const long long ns      = (long long)in_sizes[0] / LS;
  const long long nchunks = (ns + 15) / 16;

  // 1) zero the accumulator (d_out is poisoned before timing)
  long long n4 = (long long)out_size / 4;
  int zb = (int)((n4 + 255) / 256);
  gr_zero_kernel<<<zb, 256, 0, stream>>>((float4*)out, n4);

  // 2) WMMA segmented scatter-accumulate (one wave32 per 16-sample chunk)
  long long threads = nchunks * 32;
  int sb = (int)((threads + 255) / 256);
  gr_scatter_wmma_kernel<<<sb, 256, 0, stream>>>(group, weights, rays, out,
                                                 ns, nchunks);

  // 3) in-place normalize over last dim (S=8)
  long long nrows = (long long)out_size / 8;
  int nb = (int)((nrows + 255) / 256);
  gr_normalize_kernel<<<nb, 256, 0, stream>>>(out, nrows);
}